// BRGCN_2362232013294
// MI455X (gfx1250) — compile-verified
//
#include <hip/hip_runtime.h>
#include <math.h>

typedef __attribute__((ext_vector_type(2))) float v2f;
typedef __attribute__((ext_vector_type(8))) float v8f;

#define KCHUNK 256  // B-tile rows staged in LDS per iteration (16KB)

// ---------------- wave32 butterfly reductions ----------------
__device__ inline float wred_sum(float v) {
#pragma unroll
  for (int m = 16; m >= 1; m >>= 1) v += __shfl_xor(v, m, 32);
  return v;
}
__device__ inline float wred_max(float v) {
#pragma unroll
  for (int m = 16; m >= 1; m >>= 1) v = fmaxf(v, __shfl_xor(v, m, 32));
  return v;
}

// monotone float<->uint ordering for atomicMax on floats
__device__ inline unsigned f2ord(float f) {
  unsigned u = __float_as_uint(f);
  return (u & 0x80000000u) ? ~u : (u | 0x80000000u);
}
__device__ inline float ord2f(unsigned o) {
  return (o & 0x80000000u) ? __uint_as_float(o & 0x7FFFFFFFu) : __uint_as_float(~o);
}

// ---------------- WMMA fp32 GEMM: D[M,Cout] = A[M,K] @ B[K,Cout] ----------------
// AMODE 0: a(m,k) = A[m*lda + k]
// AMODE 1: k = r*zC + c ; a(m,k) = A[r*zNC + m*zC + c]   (relation-strided z)
// TRANSB 0: b(k,j) = B[k*ldb + j]
// TRANSB 1: b(k,j) = B[j*ldb + k]
// Block = 256 threads = 8 waves. Each wave owns a 32x16 output tile (2 WMMA
// accumulators); block covers 256 rows x 16 cols. B tile staged in LDS in
// KCHUNK-row chunks and shared by all waves. K-loop is wave-uniform so EXEC
// is all-ones around every WMMA.
template <int AMODE, int TRANSB>
__global__ void wmma_gemm_f32(const float* __restrict__ A, const float* __restrict__ B,
                              float* __restrict__ D, int M, int K,
                              int lda, int ldb, int ldd, int zC_shift, int zNC) {
  __shared__ float bsm[KCHUNK * 16];
  const int tid = threadIdx.x;
  const int lane = tid & 31;
  const int wave = tid >> 5;
  const int jbase = blockIdx.y * 16;
  const int jl = lane & 15;
  const int jcol = jbase + jl;
  const int kphase = (lane >> 4) << 1;  // lanes 0-15: K+0/1, lanes 16-31: K+2/3

  const int tile0 = blockIdx.x * 16 + wave * 2;  // first 16-row tile of this wave
  const int m0 = tile0 * 16 + jl;
  const int m1 = m0 + 16;
  const int ml0 = (m0 < M) ? m0 : 0;  // clamp OOB rows for loads (stores guarded)
  const int ml1 = (m1 < M) ? m1 : 0;

  v8f acc0 = {}, acc1 = {};
  for (int kbase = 0; kbase < K; kbase += KCHUNK) {
    const int kchunk = (K - kbase < KCHUNK) ? (K - kbase) : KCHUNK;
    __syncthreads();  // protect previous chunk
    for (int idx = tid; idx < kchunk * 16; idx += 256) {
      const int k = idx >> 4;
      const int j = idx & 15;
      bsm[idx] = TRANSB ? B[(size_t)(jbase + j) * ldb + (kbase + k)]
                        : B[(size_t)(kbase + k) * ldb + (jbase + j)];
    }
    __syncthreads();

    for (int k = 0; k < kchunk; k += 4) {
      const int kk = k + kphase;       // even
      const int kg = kbase + kk;       // global K index, even
      v2f a0, a1, b;
      if (AMODE == 0) {
        const float* p0 = A + (size_t)ml0 * lda + kg;
        const float* p1 = A + (size_t)ml1 * lda + kg;
        a0.x = p0[0]; a0.y = p0[1];
        a1.x = p1[0]; a1.y = p1[1];
      } else {
        const int zC = 1 << zC_shift;
        const int r = kg >> zC_shift;
        const int c = kg & (zC - 1);   // kg even => c,c+1 stay in the same row
        const float* p0 = A + (size_t)r * zNC + (size_t)ml0 * zC + c;
        const float* p1 = A + (size_t)r * zNC + (size_t)ml1 * zC + c;
        a0.x = p0[0]; a0.y = p0[1];
        a1.x = p1[0]; a1.y = p1[1];
      }
      b.x = bsm[kk * 16 + jl];
      b.y = bsm[(kk + 1) * 16 + jl];
      acc0 = __builtin_amdgcn_wmma_f32_16x16x4_f32(false, a0, false, b, (short)0, acc0,
                                                   false, false);
      acc1 = __builtin_amdgcn_wmma_f32_16x16x4_f32(false, a1, false, b, (short)0, acc1,
                                                   false, false);
    }
  }
#pragma unroll
  for (int g = 0; g < 8; ++g) {
    const int mr0 = tile0 * 16 + g + ((lane >> 4) << 3);
    if (mr0 < M) D[(size_t)mr0 * ldd + jcol] = acc0[g];
    const int mr1 = mr0 + 16;
    if (mr1 < M) D[(size_t)mr1 * ldd + jcol] = acc1[g];
  }
}

// ---------------- per-edge attention logit (wave per edge) ----------------
template <int C>
__global__ void edge_alpha_kernel(const float* __restrict__ h, const int* __restrict__ ei,
                                  const int* __restrict__ etype,
                                  const float* __restrict__ att, float* __restrict__ alpha,
                                  int Nn, int E) {
  const int lane = threadIdx.x & 31;
  const int e = blockIdx.x * 8 + (threadIdx.x >> 5);
  if (e >= E) return;
  const int src = ei[e];
  const int dst = ei[E + e];
  const int t = etype[e];
  const float* ad = att + (size_t)t * (2 * C);
  float s = 0.f;
#pragma unroll
  for (int i = 0; i < C / 32; ++i) {
    const int c = lane + i * 32;
    s += ad[c] * h[(size_t)dst * C + c];
    s += ad[C + c] * h[(size_t)src * C + c];
  }
  s = wred_sum(s);
  if (lane == 0) alpha[e] = (s > 0.f) ? s : 0.2f * s;  // leaky relu
}

// ---------------- segment max / exp-sum / normalize+scatter ----------------
__global__ void seg_max_kernel(const float* __restrict__ alpha, const int* __restrict__ ei,
                               const int* __restrict__ etype, unsigned* __restrict__ mseg,
                               int Nn, int E) {
  const int e = blockIdx.x * blockDim.x + threadIdx.x;
  if (e >= E) return;
  const int seg = etype[e] * Nn + ei[E + e];
  atomicMax(&mseg[seg], f2ord(alpha[e]));
}

__global__ void seg_expsum_kernel(float* __restrict__ alpha, const int* __restrict__ ei,
                                  const int* __restrict__ etype,
                                  const unsigned* __restrict__ mseg,
                                  float* __restrict__ denom, int Nn, int E) {
  const int e = blockIdx.x * blockDim.x + threadIdx.x;
  if (e >= E) return;
  const int seg = etype[e] * Nn + ei[E + e];
  const float v = __expf(alpha[e] - ord2f(mseg[seg]));
  alpha[e] = v;
  atomicAdd(&denom[seg], v);
}

template <int C>
__global__ void scatter_z_kernel(const float* __restrict__ h, const int* __restrict__ ei,
                                 const int* __restrict__ etype,
                                 const float* __restrict__ alpha,
                                 const float* __restrict__ denom, float* __restrict__ z,
                                 int Nn, int E) {
  const int lane = threadIdx.x & 31;
  const int e = blockIdx.x * 8 + (threadIdx.x >> 5);
  if (e >= E) return;
  const int src = ei[e];
  const int seg = etype[e] * Nn + ei[E + e];
  const float coef = alpha[e] / denom[seg];
  float* zp = z + (size_t)seg * C;
  const float* hp = h + (size_t)src * C;
#pragma unroll
  for (int i = 0; i < C / 32; ++i) {
    const int c = lane + i * 32;
    atomicAdd(&zp[c], coef * hp[c]);
  }
}

// ---------------- psi[r,n] = <z_r[n], t[n]> (wave per node) ----------------
template <int C>
__global__ void psi_dot_kernel(const float* __restrict__ z_r, const float* __restrict__ t,
                               float* __restrict__ psi_r, int Nn) {
  const int lane = threadIdx.x & 31;
  const int n = blockIdx.x * 8 + (threadIdx.x >> 5);
  if (n >= Nn) return;
  float s = 0.f;
#pragma unroll
  for (int i = 0; i < C / 32; ++i) {
    const int c = lane + i * 32;
    s += z_r[(size_t)n * C + c] * t[(size_t)n * C + c];
  }
  s = wred_sum(s);
  if (lane == 0) psi_r[n] = s;
}

// ---------------- out[n] = sum_r softmax_c(psi*vsum + base); relu or log_softmax ----
template <int C, int MODE>  // MODE 0: relu (inter-layer), 1: log_softmax (final)
__global__ void final_delta_kernel(const float* __restrict__ psi,
                                   const float* __restrict__ vsum,
                                   const float* __restrict__ base, float* __restrict__ out,
                                   int Nn, int R) {
  const int lane = threadIdx.x & 31;
  const int n = blockIdx.x * 8 + (threadIdx.x >> 5);
  if (n >= Nn) return;
  constexpr int CPL = C / 32;
  float vs[CPL], bs[CPL], acc[CPL];
#pragma unroll
  for (int i = 0; i < CPL; ++i) {
    vs[i] = vsum[(size_t)n * C + lane + i * 32];
    bs[i] = base[(size_t)n * C + lane + i * 32];
    acc[i] = 0.f;
  }
  for (int r = 0; r < R; ++r) {
    const float p = psi[(size_t)r * Nn + n];
    float val[CPL];
    float mx = -3.0e38f;
#pragma unroll
    for (int i = 0; i < CPL; ++i) {
      val[i] = p * vs[i] + bs[i];
      mx = fmaxf(mx, val[i]);
    }
    mx = wred_max(mx);
    float s = 0.f;
#pragma unroll
    for (int i = 0; i < CPL; ++i) {
      val[i] = __expf(val[i] - mx);
      s += val[i];
    }
    s = wred_sum(s);
    const float inv = 1.0f / s;
#pragma unroll
    for (int i = 0; i < CPL; ++i) acc[i] += val[i] * inv;
  }
  if (MODE == 0) {
#pragma unroll
    for (int i = 0; i < CPL; ++i)
      out[(size_t)n * C + lane + i * 32] = fmaxf(acc[i], 0.f);
  } else {
    float mx = -3.0e38f;
#pragma unroll
    for (int i = 0; i < CPL; ++i) mx = fmaxf(mx, acc[i]);
    mx = wred_max(mx);
    float s = 0.f;
#pragma unroll
    for (int i = 0; i < CPL; ++i) s += __expf(acc[i] - mx);
    s = wred_sum(s);
    const float lg = logf(s);
#pragma unroll
    for (int i = 0; i < CPL; ++i)
      out[(size_t)n * C + lane + i * 32] = (acc[i] - mx) - lg;
  }
}

// ---------------- one BRGCN layer ----------------
template <int C>
static void run_layer(const float* x_in, int Cin, const float* Wn, const float* att,
                      const float* Wq, const float* Wk, const float* Wv, const float* Ws,
                      float* out, int mode, int Nn, int E, int R, const int* ei,
                      const int* etype, float* h, float* z, float* tbuf, float* ksum,
                      float* vsum, float* base, float* alpha, unsigned* mseg, float* denom,
                      float* psi, hipStream_t stream) {
  const dim3 blk(256);
  const int mtiles = (Nn + 15) / 16;
  const dim3 grdN((mtiles + 15) / 16, C / 16);  // 16 row-tiles (256 rows) per block
  const int shift = (C == 128) ? 7 : 6;

  // h = x_in @ Wn
  wmma_gemm_f32<0, 0><<<grdN, blk, 0, stream>>>(x_in, Wn, h, Nn, Cin, Cin, C, C, 0, 0);

  hipMemsetAsync(z, 0, (size_t)R * Nn * C * sizeof(float), stream);
  hipMemsetAsync(mseg, 0, (size_t)R * Nn * sizeof(unsigned), stream);
  hipMemsetAsync(denom, 0, (size_t)R * Nn * sizeof(float), stream);

  edge_alpha_kernel<C><<<(E + 7) / 8, blk, 0, stream>>>(h, ei, etype, att, alpha, Nn, E);
  seg_max_kernel<<<(E + 255) / 256, blk, 0, stream>>>(alpha, ei, etype, mseg, Nn, E);
  seg_expsum_kernel<<<(E + 255) / 256, blk, 0, stream>>>(alpha, ei, etype, mseg, denom, Nn, E);
  scatter_z_kernel<C><<<(E + 7) / 8, blk, 0, stream>>>(h, ei, etype, alpha, denom, z, Nn, E);

  // ksum = sum_r z_r @ Wk_r ; vsum = sum_r z_r @ Wv_r  (single GEMM each, K = R*C)
  wmma_gemm_f32<1, 0><<<grdN, blk, 0, stream>>>(z, Wk, ksum, Nn, R * C, 0, C, C, shift, Nn * C);
  wmma_gemm_f32<1, 0><<<grdN, blk, 0, stream>>>(z, Wv, vsum, Nn, R * C, 0, C, C, shift, Nn * C);
  // base = h @ Ws
  wmma_gemm_f32<0, 0><<<grdN, blk, 0, stream>>>(h, Ws, base, Nn, C, C, C, C, 0, 0);

  // psi[r,n] = < z_r[n], ksum[n] @ Wq_r^T >
  for (int r = 0; r < R; ++r) {
    wmma_gemm_f32<0, 1><<<grdN, blk, 0, stream>>>(ksum, Wq + (size_t)r * C * C, tbuf, Nn, C,
                                                  C, C, C, 0, 0);
    psi_dot_kernel<C><<<(Nn + 7) / 8, blk, 0, stream>>>(z + (size_t)r * Nn * C, tbuf,
                                                        psi + (size_t)r * Nn, Nn);
  }

  if (mode == 0)
    final_delta_kernel<C, 0><<<(Nn + 7) / 8, blk, 0, stream>>>(psi, vsum, base, out, Nn, R);
  else
    final_delta_kernel<C, 1><<<(Nn + 7) / 8, blk, 0, stream>>>(psi, vsum, base, out, Nn, R);
}

extern "C" void kernel_launch(void* const* d_in, const int* in_sizes, int n_in,
                              void* d_out, int out_size, void* d_ws, size_t ws_size,
                              hipStream_t stream) {
  const float* x    = (const float*)d_in[0];
  const int* ei     = (const int*)d_in[1];
  const int* etype  = (const int*)d_in[2];
  const float* Wn1  = (const float*)d_in[3];
  const float* att1 = (const float*)d_in[4];
  const float* Wq1  = (const float*)d_in[5];
  const float* Wk1  = (const float*)d_in[6];
  const float* Wv1  = (const float*)d_in[7];
  const float* Ws1  = (const float*)d_in[8];
  const float* Wn2  = (const float*)d_in[9];
  const float* att2 = (const float*)d_in[10];
  const float* Wq2  = (const float*)d_in[11];
  const float* Wk2  = (const float*)d_in[12];
  const float* Wv2  = (const float*)d_in[13];
  const float* Ws2  = (const float*)d_in[14];

  const int IN_C = 128, HID = 128, OUT = 64, R = 8;
  const int Nn = in_sizes[0] / IN_C;
  const int E  = in_sizes[1] / 2;

  // workspace layout (floats); z sized for layer 1 (C=128), reused for layer 2
  float* ws = (float*)d_ws;
  float* h     = ws;  ws += (size_t)Nn * 128;
  float* z     = ws;  ws += (size_t)R * Nn * 128;
  float* tbuf  = ws;  ws += (size_t)Nn * 128;
  float* ksum  = ws;  ws += (size_t)Nn * 128;
  float* vsum  = ws;  ws += (size_t)Nn * 128;
  float* base  = ws;  ws += (size_t)Nn * 128;
  float* hmid  = ws;  ws += (size_t)Nn * 128;
  float* alpha = ws;  ws += (size_t)E;
  unsigned* mseg = (unsigned*)ws;  ws += (size_t)R * Nn;
  float* denom = ws;  ws += (size_t)R * Nn;
  float* psi   = ws;  ws += (size_t)R * Nn;

  // layer 1: 128 -> 128, relu
  run_layer<128>(x, IN_C, Wn1, att1, Wq1, Wk1, Wv1, Ws1, hmid, /*mode=*/0, Nn, E, R, ei,
                 etype, h, z, tbuf, ksum, vsum, base, alpha, mseg, denom, psi, stream);
  // layer 2: 128 -> 64, log_softmax into d_out
  run_layer<64>(hmid, HID, Wn2, att2, Wq2, Wk2, Wv2, Ws2, (float*)d_out, /*mode=*/1, Nn, E,
                R, ei, etype, h, z, tbuf, ksum, vsum, base, alpha, mseg, denom, psi, stream);

  (void)n_in; (void)out_size; (void)ws_size; (void)OUT;
}